// MarginalSampler_35828617183721
// MI455X (gfx1250) — compile-verified
//
#include <hip/hip_runtime.h>
#include <math.h>

typedef float v2f __attribute__((ext_vector_type(2)));
typedef float v8f __attribute__((ext_vector_type(8)));
typedef float f4  __attribute__((ext_vector_type(4)));
typedef int   v4i __attribute__((vector_size(16)));   // matches builtin's param type

#define SEG 64
#define DK  32
#define DW  64
#define SV  36   // padded LDS stride for V rows (floats): 16 distinct banks, 16B aligned
#define SA  65   // padded LDS stride for A (64x64 Gram / Cholesky factor)

#if __has_builtin(__builtin_amdgcn_global_load_async_to_lds_b128)
#define HAVE_ASYNC 1
#else
#define HAVE_ASYNC 0
#endif

__device__ __forceinline__ void wait_async0() {
#if __has_builtin(__builtin_amdgcn_s_wait_asynccnt)
    __builtin_amdgcn_s_wait_asynccnt(0);
#else
    asm volatile("s_wait_asynccnt 0x0" ::: "memory");
#endif
}

__global__ __launch_bounds__(64) void dpp_marginal_kernel(
    const float* __restrict__ kernels,
    const float* __restrict__ words,
    float* __restrict__ out)
{
    __shared__ float Vs[SEG * SV];   // 9.2 KB  V tile (64x32, padded)
    __shared__ float As[SEG * SA];   // 16.6 KB L then Cholesky factor (lower)
    __shared__ float Xs[SEG * SEG];  // 16 KB   per-thread solve columns
    __shared__ float Dg[SEG];        // sqrt of Cholesky diagonal
    __shared__ float Ms[SEG];        // marginals

    const int t    = threadIdx.x;    // 0..63
    const int lane = t & 31;
    const int wave = t >> 5;
    const int b    = blockIdx.x;

    const float* Vg = kernels + (size_t)b * SEG * DK;
    const float* Wg = words   + (size_t)b * SEG * DW;
    float*       Og = out     + (size_t)b * SEG * DW;

    // ---- stage V (64x32 fp32) into padded LDS: 512 x 16B, 8 per thread ----
#if HAVE_ASYNC
    #pragma unroll
    for (int i = 0; i < 8; ++i) {
        int q    = i * 64 + t;        // 16B chunk index (coalesced)
        int row  = q >> 3;            // 8 chunks per 32-float row
        int col4 = (q & 7) * 4;
        __builtin_amdgcn_global_load_async_to_lds_b128(
            (v4i*)(Vg + q * 4),
            (__attribute__((address_space(3))) v4i*)&Vs[row * SV + col4],
            0, 0);
    }
    wait_async0();
#else
    #pragma unroll
    for (int i = 0; i < 8; ++i) {
        int q    = i * 64 + t;
        int row  = q >> 3;
        int col4 = (q & 7) * 4;
        f4 v = ((const f4*)Vg)[q];
        *(f4*)&Vs[row * SV + col4] = v;
    }
#endif
    __syncthreads();

    // ---- lower triangle of L = V V^T via V_WMMA_F32_16X16X4_F32 ----
    // 10 lower 16x16 tiles; 5 per wave. Upper triangle is never read.
    const int m16 = lane & 15;        // matrix row/col within tile
    const int kh  = lane >> 4;        // K half-select per ISA A layout
    for (int s5 = 0; s5 < 5; ++s5) {
        int tile = wave * 5 + s5;     // 0..9 (lower-tri tile id)
        int mi = (tile >= 6) ? 3 : (tile >= 3) ? 2 : (tile >= 1) ? 1 : 0;
        int ni = tile - ((mi * (mi + 1)) >> 1);
        int tm = mi * 16, tn = ni * 16;
        v8f acc = {0.f, 0.f, 0.f, 0.f, 0.f, 0.f, 0.f, 0.f};
        #pragma unroll
        for (int kb = 0; kb < DK; kb += 4) {
            // A: 16x4 slice of V rows [tm..]; lanes 0-15 K=kb+0/1, 16-31 K=kb+2/3
            const float* ap = &Vs[(tm + m16) * SV + kb + 2 * kh];
            // B: 4x16 slice of V^T == rows [tn..] of V, mirrored layout
            const float* bp = &Vs[(tn + m16) * SV + kb + 2 * kh];
            v2f a;  a.x  = ap[0]; a.y  = ap[1];
            v2f bb; bb.x = bp[0]; bb.y = bp[1];
            acc = __builtin_amdgcn_wmma_f32_16x16x4_f32(
                      false, a, false, bb, (short)0, acc, false, false);
        }
        // D layout: VGPR r -> row (r + 8*kh), col = lane&15
        #pragma unroll
        for (int r = 0; r < 8; ++r)
            As[(tm + r + 8 * kh) * SA + tn + m16] = acc[r];
    }
    __syncthreads();

    // ---- A = L + I ----
    As[t * SA + t] += 1.0f;
    __syncthreads();

    // ---- Cholesky A = C C^T (lower), thread t owns row t; SPD => no pivot.
    // sqrt(diag) kept in Dg so As[k][k] is never overwritten => 2 barriers/step.
    for (int k = 0; k < SEG; ++k) {
        float rs = 1.0f / sqrtf(As[k * SA + k]);  // As[k][k] final after loop-end barrier
        if (t == k) Dg[k] = As[k * SA + k] * rs;
        if (t > k)  As[t * SA + k] *= rs;         // scale column k (strictly-lower)
        __syncthreads();
        if (t > k) {
            float aik = As[t * SA + k];
            for (int j = k + 1; j <= t; ++j)      // trailing update, row t
                As[t * SA + j] -= aik * As[j * SA + k];
        }
        __syncthreads();
    }

    // ---- diag(A^-1)_t = sum_i (C^-1)_{i,t}^2 : forward-solve C x = e_t ----
    // Thread-private LDS column Xs[i*64 + t] (fixed bank t) => no barriers.
    float ssq = 0.0f;
    for (int i = t; i < SEG; ++i) {
        float s = (i == t) ? 1.0f : 0.0f;
        for (int k = t; k < i; ++k)
            s -= As[i * SA + k] * Xs[k * SEG + t];
        float xi = s / Dg[i];
        Xs[i * SEG + t] = xi;
        ssq += xi * xi;
    }
    Ms[t] = 1.0f - ssq;
    __syncthreads();

    // ---- out = diag(marginal) @ W : coalesced 16B, non-temporal stream ----
    const f4* W4 = (const f4*)Wg;
    f4*       O4 = (f4*)Og;
    #pragma unroll
    for (int i = 0; i < 16; ++i) {
        int q = i * 64 + t;           // 16B index; row = q >> 4
        float m = Ms[q >> 4];
        f4 w = __builtin_nontemporal_load(&W4[q]);
        w *= m;
        __builtin_nontemporal_store(w, &O4[q]);
    }
}

extern "C" void kernel_launch(void* const* d_in, const int* in_sizes, int n_in,
                              void* d_out, int out_size, void* d_ws, size_t ws_size,
                              hipStream_t stream) {
    (void)n_in; (void)d_ws; (void)ws_size; (void)out_size;
    const float* kernels = (const float*)d_in[0];
    const float* words   = (const float*)d_in[1];
    float*       out     = (float*)d_out;

    const int T = in_sizes[0] / DK;   // 524288
    const int B = T / SEG;            // 8192 segments -> one block each
    dpp_marginal_kernel<<<B, 64, 0, stream>>>(kernels, words, out);
}